// Net_5634997092537
// MI455X (gfx1250) — compile-verified
//
#include <hip/hip_runtime.h>
#include <hip/hip_bf16.h>

// ---------------------------------------------------------------------------
// SPDNet forward for MI455X (gfx1250, wave32, WMMA, TDM).
//  - rectify layers elided (eigen-interlacing: lambda_min >= 1 >> 1e-4)
//  - bilinear transforms as f32 WMMA (v_wmma_f32_16x16x4_f32)
//  - A and B tiles double-buffered in LDS via the Tensor Data Mover
//    (tensor_load_to_lds): 1 DMA per operand tile, HW zero-fill at ragged
//    edges, HW LDS padding (pad 8 dw / 64 dw) giving the 72-float row stride
//    that keeps the upper half-wave's K+2 row on disjoint banks.
//  - final 50x50 matrix log via parallel cyclic Jacobi in LDS
// ---------------------------------------------------------------------------

typedef float v2f __attribute__((ext_vector_type(2)));
typedef float v8f __attribute__((ext_vector_type(8)));
typedef unsigned int v4u __attribute__((ext_vector_type(4)));
typedef int v4i __attribute__((ext_vector_type(4)));
typedef int v8i __attribute__((ext_vector_type(8)));

#define KT     32      // K tile staged in LDS
#define LDT    72      // padded LDS row stride (floats) = 64 + 8 (TDM pad)

// Issue one TDM 2D tile load: KT x 64 f32 tile at (row pitch = rowStride
// floats) into LDS with +8dw pad every 64dw. cols/rows = valid extent from
// the tile origin (beyond that the TDM zero-fills).
__device__ __forceinline__ void tdm_load_tile(const float* g, unsigned lds,
                                              int rows, int cols, int rowStride)
{
    unsigned long long ga = (unsigned long long)g;
    v4u g0;
    g0[0] = 1u;                                   // count=1, user descriptor
    g0[1] = lds;                                  // LDS byte address
    g0[2] = (unsigned)(ga & 0xFFFFFFFFu);         // global_addr[31:0]
    g0[3] = (unsigned)((ga >> 32) & 0x01FFFFFFu)  // global_addr[56:32]
          | (2u << 30);                           // type = 2 ("image")
    v8i g1;
    g1[0] = (2 << 16)      // data_size = 4B
          | (1 << 20)      // pad_enable
          | (5 << 22)      // pad_interval: every 64 DWORDs
          | (7 << 25);     // pad_amount: 8 DWORDs
    g1[1] = (cols & 0xFFFF) << 16;                         // tensor_dim0 lo
    g1[2] = ((cols >> 16) & 0xFFFF) | ((rows & 0xFFFF) << 16); // dim0 hi|dim1 lo
    g1[3] = ((rows >> 16) & 0xFFFF) | (64 << 16);          // dim1 hi | tile_dim0=64
    g1[4] = KT;                                            // tile_dim1=KT, tile_dim2=0
    g1[5] = rowStride;                                     // tensor_dim0_stride lo
    g1[6] = 0;                                             // stride hi | dim1_stride lo
    g1[7] = 0;
    v4i gz = {0, 0, 0, 0};
#if defined(__clang_major__) && (__clang_major__ >= 23)
    v8i gz8 = {0, 0, 0, 0, 0, 0, 0, 0};
    __builtin_amdgcn_tensor_load_to_lds(g0, g1, gz, gz, gz8, 0);
#else
    __builtin_amdgcn_tensor_load_to_lds(g0, g1, gz, gz, 0);
#endif
}

// C[M,N] = sum_k A[k,m] * B[k,n]   (A^T * B, both K-major), batched.
// 128-thread block = 4 waves; each wave computes a 16(M) x 64(N) strip with
// 4 f32 WMMA accumulators. Operand tiles staged in LDS by the TDM.
__global__ void __launch_bounds__(128)
spd_gemm_AtB(const float* __restrict__ A, const float* __restrict__ B,
             float* __restrict__ C,
             int K, int M, int N,
             long strideA, long strideB, long strideC,
             int lda, int ldb, int ldc)
{
    __shared__ float Asm[2][KT][LDT];
    __shared__ float Bsm[2][KT][LDT];

    const int tid  = threadIdx.x;
    const int wave = tid >> 5;
    const int lane = tid & 31;
    const int half = lane >> 4;   // 0: lanes 0-15, 1: lanes 16-31
    const int l16  = lane & 15;

    const int b     = blockIdx.z;
    const int nTile = blockIdx.x * 64;
    const int mTile = blockIdx.y * 64;

    A += (long)b * strideA;
    B += (long)b * strideB;
    C += (long)b * strideC;

    const unsigned ldsA0 = (unsigned)(unsigned long long)&Asm[0][0][0];
    const unsigned ldsA1 = (unsigned)(unsigned long long)&Asm[1][0][0];
    const unsigned ldsB0 = (unsigned)(unsigned long long)&Bsm[0][0][0];
    const unsigned ldsB1 = (unsigned)(unsigned long long)&Bsm[1][0][0];

    // Issue both operand tiles for K-block starting at k0 into buffer bufsel.
    auto issue_tile = [&](int k0, int bufsel) {
        tdm_load_tile(A + (long)k0 * lda + mTile,
                      bufsel ? ldsA1 : ldsA0, K - k0, M - mTile, lda);
        tdm_load_tile(B + (long)k0 * ldb + nTile,
                      bufsel ? ldsB1 : ldsB0, K - k0, N - nTile, ldb);
    };

    v8f acc[4];
    const v8f z = {0.f,0.f,0.f,0.f,0.f,0.f,0.f,0.f};
#pragma unroll
    for (int t = 0; t < 4; ++t) acc[t] = z;

    const int nTiles = (K + KT - 1) / KT;
    if (wave == 0) issue_tile(0, 0);

    for (int t = 0; t < nTiles; ++t) {
        const int buf = t & 1;
        if (t + 1 < nTiles) {
            if (wave == 0) issue_tile((t + 1) * KT, buf ^ 1);
            // TDM completes in order per wave: <=2 outstanding => tile t done
            __builtin_amdgcn_s_wait_tensorcnt(2);
        } else {
            __builtin_amdgcn_s_wait_tensorcnt(0);
        }
        __syncthreads();

        const int kbase = t * KT;
        int kt = K - kbase; if (kt > KT) kt = KT;   // K is a multiple of 4

        for (int kk = 0; kk < kt; kk += 4) {
            // fragment rows: VGPR0 = K+0 (lanes 0-15) / K+2 (lanes 16-31),
            //                VGPR1 = K+1 / K+3.
            const int kr = kk + 2 * half;

            const float* ap = &Asm[buf][kr][wave * 16 + l16];
            v2f afrag;
            afrag.x = ap[0];
            afrag.y = ap[LDT];      // row kr+1

            const float* bp = &Bsm[buf][kr][l16];
#pragma unroll
            for (int u = 0; u < 4; ++u) {
                v2f bfrag;
                bfrag.x = bp[u * 16];
                bfrag.y = bp[u * 16 + LDT];
                acc[u] = __builtin_amdgcn_wmma_f32_16x16x4_f32(
                    /*neg_a=*/false, afrag, /*neg_b=*/false, bfrag,
                    /*c_mod=*/(short)0, acc[u],
                    /*reuse_a=*/false, /*reuse_b=*/false);
            }
        }
        __syncthreads();   // all waves done reading before buffer is reused
    }

    // C layout: acc element r -> row (r + 8*half), col = lane&15 within tile.
    const int mBase = mTile + wave * 16;
    if ((mBase + 15 < M) && (nTile + 63 < N)) {
        // interior fast path: no bounds checks
#pragma unroll
        for (int u = 0; u < 4; ++u) {
            const int n = nTile + u * 16 + l16;
#pragma unroll
            for (int r = 0; r < 8; ++r) {
                const int m = mBase + r + 8 * half;
                C[(long)m * ldc + n] = acc[u][r];
            }
        }
    } else {
#pragma unroll
        for (int u = 0; u < 4; ++u) {
            const int n = nTile + u * 16 + l16;
            if (n < N) {
#pragma unroll
                for (int r = 0; r < 8; ++r) {
                    const int m = mBase + r + 8 * half;
                    if (m < M) C[(long)m * ldc + n] = acc[u][r];
                }
            }
        }
    }
}

// ---------------------------------------------------------------------------
// Batched 50x50 symmetric eigensolver (parallel cyclic Jacobi, fully in LDS),
// then tangent-space log-map fused with the 1275 -> 7 linear head.
// One 256-thread workgroup per batch element.
// ---------------------------------------------------------------------------
#define NT    50
#define NPAD  52
#define NPAIR (NT / 2)
#define NFEAT 1275
#define SWEEPS 10

__global__ void __launch_bounds__(256)
spd_log_tangent(const float* __restrict__ Y,   // [B,50,50]
                const float* __restrict__ Wl,  // [7,1275]
                const float* __restrict__ bl,  // [7]
                float* __restrict__ out)       // [B,7]
{
    __shared__ float Am[NT][NPAD];
    __shared__ float Vm[NT][NPAD];
    __shared__ float cs[NPAIR], sn[NPAIR];
    __shared__ int   pp[NPAIR], qq[NPAIR];
    __shared__ float lw[NT];
    __shared__ float red[256];

    const int tid = threadIdx.x;
    const int b   = blockIdx.x;
    const float* Yb = Y + (long)b * NT * NT;

    for (int idx = tid; idx < NT * NT; idx += 256) {
        int i = idx / NT, j = idx - i * NT;
        Am[i][j] = Yb[idx];
        Vm[i][j] = (i == j) ? 1.f : 0.f;
    }
    __syncthreads();

    for (int sweep = 0; sweep < SWEEPS; ++sweep) {
        for (int r = 0; r < NT - 1; ++r) {
            // Round-robin tournament: 25 disjoint (p,q) pairs this round.
            if (tid < NPAIR) {
                int j  = tid;
                int a  = (j == 0) ? 0 : ((r + j - 1) % (NT - 1)) + 1;
                int b2 = ((r + (NT - 1 - j) - 1) % (NT - 1)) + 1;
                int p = a < b2 ? a : b2;
                int q = a < b2 ? b2 : a;
                pp[j] = p; qq[j] = q;
                float app = Am[p][p], aqq = Am[q][q], apq = Am[p][q];
                float c = 1.f, s = 0.f;
                if (fabsf(apq) > 1e-12f) {
                    float tau = (aqq - app) / (2.f * apq);
                    float tt  = (tau >= 0.f ? 1.f : -1.f) /
                                (fabsf(tau) + sqrtf(1.f + tau * tau));
                    c = 1.f / sqrtf(1.f + tt * tt);
                    s = tt * c;
                }
                cs[j] = c; sn[j] = s;
            }
            __syncthreads();
            // A <- G^T A   (rows p,q; pairs are disjoint -> parallel-safe)
            for (int idx = tid; idx < NPAIR * NT; idx += 256) {
                int j = idx / NT, col = idx - j * NT;
                int p = pp[j], q = qq[j];
                float c = cs[j], s = sn[j];
                float ap = Am[p][col], aq = Am[q][col];
                Am[p][col] = c * ap - s * aq;
                Am[q][col] = s * ap + c * aq;
            }
            __syncthreads();
            // A <- A G ; V <- V G   (cols p,q)
            for (int idx = tid; idx < NPAIR * NT; idx += 256) {
                int j = idx / NT, row = idx - j * NT;
                int p = pp[j], q = qq[j];
                float c = cs[j], s = sn[j];
                float ap = Am[row][p], aq = Am[row][q];
                Am[row][p] = c * ap - s * aq;
                Am[row][q] = s * ap + c * aq;
                float vp = Vm[row][p], vq = Vm[row][q];
                Vm[row][p] = c * vp - s * vq;
                Vm[row][q] = s * vp + c * vq;
            }
            __syncthreads();
        }
    }

    if (tid < NT) lw[tid] = logf(fmaxf(Am[tid][tid], 1e-4f));
    __syncthreads();

    // feat[f] = coef * (V diag(lw) V^T)[i,j], fused into out = feat @ Wl^T + bl
    float partial[7] = {0.f, 0.f, 0.f, 0.f, 0.f, 0.f, 0.f};
    const float SQRT2 = 1.41421356237309515f;
    for (int f = tid; f < NFEAT; f += 256) {
        int i = 0, rem = f;                 // row-major upper-triangle index
        while (rem >= NT - i) { rem -= NT - i; ++i; }
        int j = i + rem;
        float sij = 0.f;
        for (int k = 0; k < NT; ++k) sij += Vm[i][k] * lw[k] * Vm[j][k];
        float val = sij * ((i == j) ? 1.f : SQRT2);
#pragma unroll
        for (int c = 0; c < 7; ++c) partial[c] += val * Wl[c * NFEAT + f];
    }
    for (int c = 0; c < 7; ++c) {
        __syncthreads();
        red[tid] = partial[c];
        __syncthreads();
        for (int off = 128; off > 0; off >>= 1) {
            if (tid < off) red[tid] += red[tid + off];
            __syncthreads();
        }
        if (tid == 0) out[b * 7 + c] = red[0] + bl[c];
    }
}

// ---------------------------------------------------------------------------
// Host-side launcher.
// Inputs: x[256,400,400], W1[400,200], W2[200,100], W3[100,50],
//         Wl[7,1275], bl[7]  -> out[256,7]
// Both GEMM halves use the A^T*B form (X and all Y stages are symmetric).
// ---------------------------------------------------------------------------
extern "C" void kernel_launch(void* const* d_in, const int* in_sizes, int n_in,
                              void* d_out, int out_size, void* d_ws, size_t ws_size,
                              hipStream_t stream)
{
    const float* x  = (const float*)d_in[0];
    const float* W1 = (const float*)d_in[1];
    const float* W2 = (const float*)d_in[2];
    const float* W3 = (const float*)d_in[3];
    const float* Wl = (const float*)d_in[4];
    const float* bl = (const float*)d_in[5];
    float* out = (float*)d_out;

    // Workspace layout (floats). T buffer reused across all three stages.
    float* T  = (float*)d_ws;            // max 256*400*200 = 20,480,000
    float* Y1 = T  + 20480000;           // 256*200*200    = 10,240,000
    float* Y2 = Y1 + 10240000;           // 256*100*100    =  2,560,000
    float* Y3 = Y2 + 2560000;            // 256* 50* 50    =    640,000
                                         // total ~135.7 MB

    const int B = 256;
    dim3 blk(128);
    auto cdiv = [](int a, int b) { return (a + b - 1) / b; };

    // Stage 1: n=400 -> m=200.  T = X^T W1 (== X W1), Y1 = W1^T T
    spd_gemm_AtB<<<dim3(cdiv(200,64), cdiv(400,64), B), blk, 0, stream>>>(
        x, W1, T, 400, 400, 200, 400L*400, 0, 400L*200, 400, 200, 200);
    spd_gemm_AtB<<<dim3(cdiv(200,64), cdiv(200,64), B), blk, 0, stream>>>(
        W1, T, Y1, 400, 200, 200, 0, 400L*200, 200L*200, 200, 200, 200);

    // Stage 2: n=200 -> m=100.
    spd_gemm_AtB<<<dim3(cdiv(100,64), cdiv(200,64), B), blk, 0, stream>>>(
        Y1, W2, T, 200, 200, 100, 200L*200, 0, 200L*100, 200, 100, 100);
    spd_gemm_AtB<<<dim3(cdiv(100,64), cdiv(100,64), B), blk, 0, stream>>>(
        W2, T, Y2, 200, 100, 100, 0, 200L*100, 100L*100, 100, 100, 100);

    // Stage 3: n=100 -> m=50.
    spd_gemm_AtB<<<dim3(cdiv(50,64), cdiv(100,64), B), blk, 0, stream>>>(
        Y2, W3, T, 100, 100, 50, 100L*100, 0, 100L*50, 100, 50, 50);
    spd_gemm_AtB<<<dim3(cdiv(50,64), cdiv(50,64), B), blk, 0, stream>>>(
        W3, T, Y3, 100, 50, 50, 0, 100L*50, 50L*50, 50, 50, 50);

    // Final: eigen-log tangent map + linear head.
    spd_log_tangent<<<dim3(B), dim3(256), 0, stream>>>(Y3, Wl, bl, out);
}